// Conv_input_59631325938010
// MI455X (gfx1250) — compile-verified
//
#include <hip/hip_runtime.h>
#include <hip/hip_bf16.h>
#include <stdint.h>

// Causal 7-tap sliding window gather on gfx1250 (MI455X).
//   out[b, i, j, c] = x[b, i - (6 - j), c], zero when index < 0.
// Pure data movement (0 FLOPs): 32.8 MB in + 229.4 MB out => ~11 us HBM floor.
// CDNA5 path: stage each input row into LDS exactly once via
// global_load_async_to_lds_b128, fan it out to its 7 output slots with
// global_store_async_from_lds_b128 (no VGPR data round-trip, no 7x L2 read
// amplification). Dependencies tracked with ASYNCcnt.

#define B_    32
#define S_    2000
#define C_    128
#define TAPS_ 7
#define HALO_ (TAPS_ - 1)           // 6
#define TILE_ 40                    // 2000 / 40 = 50 tiles per batch
#define LROWS_ (TILE_ + HALO_)      // 46 rows staged in LDS
#define C4_   (C_ / 4)              // 32 float4 chunks per row
#define LOAD_CHUNKS_  (LROWS_ * C4_)         // 1472
#define STORE_CHUNKS_ (TILE_ * TAPS_ * C4_)  // 8960 (=> 35 iters/thread, exact)
#define NTHREADS_ 256
#define NWAVES_   (NTHREADS_ / 32)  // 8

__device__ __forceinline__ void async_load_b128(const float* gaddr, uint32_t lds_off) {
    // copies 16 bytes from global address gaddr to LDS offset lds_off   (per active lane), ASYNCcnt++
    asm volatile("global_load_async_to_lds_b128 %0, %1, off"
                 :
                 : "v"(lds_off), "v"(gaddr)
                 : "memory");
}

__device__ __forceinline__ void async_store_b128(float* gaddr, uint32_t lds_off) {
    // copies 16 bytes from LDS offset lds_off to global address gaddr   (per active lane), ASYNCcnt++
    asm volatile("global_store_async_from_lds_b128 %0, %1, off"
                 :
                 : "v"(gaddr), "v"(lds_off)
                 : "memory");
}

__device__ __forceinline__ void wait_async_zero() {
    asm volatile("s_wait_asynccnt 0" ::: "memory");
}

__global__ __launch_bounds__(NTHREADS_)
void window7_async_kernel(const float* __restrict__ x, float* __restrict__ out) {
    __shared__ float lds[LROWS_ * C_];   // 46 * 128 * 4 = 23,552 bytes

    const int wg   = blockIdx.x;
    const int b    = wg / (S_ / TILE_);
    const int tile = wg % (S_ / TILE_);
    const int t0   = tile * TILE_;
    const int tid  = threadIdx.x;

    // Low 32 bits of the flat pointer to a __shared__ object are the LDS byte
    // offset within this workgroup's allocation (ISA 10.2 aperture mapping).
    const uint32_t lds_base = (uint32_t)(uintptr_t)(void*)lds;

    const float* xb = x + (size_t)b * S_ * C_;

    // ---- Phase 1: stage rows [t0-6, t0+TILE) into LDS rows [0, LROWS) ----
    // Chunk q is linear in both spaces: LDS byte q*16, global byte
    // (t0-6)*512 + q*16. Only the first tile has the causal zero pad
    // (chunks 0..191 = rows -6..-1), handled in a uniform branch.
    if (t0 == 0) {
        if (tid < HALO_ * C4_) {                    // 192 pad chunks
            float4 z = {0.0f, 0.0f, 0.0f, 0.0f};
            *reinterpret_cast<float4*>(
                reinterpret_cast<char*>(lds) + (size_t)tid * 16) = z;
        } else {
            async_load_b128(xb + (tid * 4 - HALO_ * C_),
                            lds_base + (uint32_t)(tid * 16));
        }
        for (int q = tid + NTHREADS_; q < LOAD_CHUNKS_; q += NTHREADS_) {
            async_load_b128(xb + (q * 4 - HALO_ * C_),
                            lds_base + (uint32_t)(q * 16));
        }
    } else {
        const float* gb = xb + (size_t)(t0 - HALO_) * C_;
        for (int q = tid; q < LOAD_CHUNKS_; q += NTHREADS_) {
            async_load_b128(gb + q * 4, lds_base + (uint32_t)(q * 16));
        }
    }
    wait_async_zero();   // my async loads landed in LDS
    __syncthreads();     // everyone's loads + zero-fills visible

    // ---- Phase 2: fan out to the 7 taps straight from LDS ----
    // Chunk index rt = il*7 + j; out[b, t0+il, j, :] = LDS row (il + j).
    // Global address is linear in rt (stride 512 B); LDS row is tracked
    // incrementally: rt += 8  <=>  il += 1, j += 1 (wrap j at 7, il += 1).
    const int c4 = tid & 31;
    const int w  = tid >> 5;                       // wave id 0..7
    int j  = (w >= TAPS_) ? (w - TAPS_) : w;
    int il = (w >= TAPS_) ? 1 : 0;

    float* g = out + (size_t)b * S_ * TAPS_ * C_
                   + (size_t)(t0 * TAPS_ + w) * C_ + (size_t)c4 * 4;
    uint32_t lbase = lds_base + (uint32_t)(c4 * 16);

#pragma unroll 7
    for (int rt = w; rt < TILE_ * TAPS_; rt += NWAVES_) {
        async_store_b128(g, lbase + (uint32_t)((il + j) * (C4_ * 16)));
        g += NWAVES_ * C_;                         // +4096 B
        j += 1; il += 1;
        if (j >= TAPS_) { j -= TAPS_; il += 1; }   // 8 = 7 + 1 carry
    }
    wait_async_zero();   // drain before LDS dealloc (s_endpgm also waits idle)
}

extern "C" void kernel_launch(void* const* d_in, const int* in_sizes, int n_in,
                              void* d_out, int out_size, void* d_ws, size_t ws_size,
                              hipStream_t stream) {
    (void)in_sizes; (void)n_in; (void)out_size; (void)d_ws; (void)ws_size;
    const float* x = (const float*)d_in[0];
    float* out = (float*)d_out;
    dim3 grid(B_ * (S_ / TILE_));   // 32 * 50 = 1600 workgroups
    dim3 block(NTHREADS_);
    window7_async_kernel<<<grid, block, 0, stream>>>(x, out);
}